// ConvEnhanced_65481071396626
// MI455X (gfx1250) — compile-verified
//
#include <hip/hip_runtime.h>

typedef float v2f __attribute__((ext_vector_type(2)));
typedef float v8f __attribute__((ext_vector_type(8)));

__device__ __forceinline__ float fast_sigmoid(float t) {
    // sigmoid(t) = 1 / (1 + exp(-t)); v_exp_f32 + v_rcp_f32
    float e = __expf(-t);
    return __builtin_amdgcn_rcpf(1.0f + e);
}

// One wave32 handles a tile of 16 batch elements (16x16 f32 = 1KB of x).
//   lane l: n = l&15 (batch row in tile), hi = l>>4 (K-pair half)
// WMMA mapping (V_WMMA_F32_16X16X4_F32, chained K=4 slices -> K=16):
//   A[m,k] = conv_w[k]  (broadcast rows)  : lane holds w[4*it + 2*hi + {0,1}]
//   B[k,n] = x[n][k]                      : lane holds x[n][4*it + 2*hi + {0,1}]
//   D[m,n] = dot(w, x[n])  -> acc[0] on every lane = dot for batch n
__global__ __launch_bounds__(256)
void ConvEnhanced_fused_kernel(const float* __restrict__ x,
                               const float* __restrict__ conv_w,
                               const float* __restrict__ conv_b,
                               const float* __restrict__ attn_w,
                               const float* __restrict__ alpha_p,
                               float* __restrict__ out,
                               int batch)
{
    const int lane = threadIdx.x & 31;
    const int n    = lane & 15;
    const int hi   = lane >> 4;

    const int waveId = (int)((blockIdx.x * blockDim.x + threadIdx.x) >> 5);
    const int nWaves = (int)((gridDim.x * blockDim.x) >> 5);
    const int nTiles = (batch + 15) >> 4;

    // Per-lane slice of the 16 conv weights (tiny; hits constant/L2 cache).
    v2f wA[4];
#pragma unroll
    for (int it = 0; it < 4; ++it)
        wA[it] = *(const v2f*)(conv_w + it * 4 + hi * 2);

    const float b0 = conv_b[0];
    const float aw = attn_w[0];
    const float al = alpha_p[0];
    const float PI = 3.14159265358979323846f;

    for (int tile = waveId; tile < nTiles; tile += nWaves) {
        int row  = tile * 16 + n;
        int rowc = (row < batch) ? row : (batch - 1);   // clamp so EXEC stays all-1s for WMMA
        const float* p = x + (size_t)rowc * 16 + hi * 2;

        // Streaming (use-once) data: non-temporal 8B loads; wave covers full cachelines.
        v2f xb[4];
#pragma unroll
        for (int it = 0; it < 4; ++it)
            xb[it] = __builtin_nontemporal_load((const v2f*)(p + it * 4));

        // dot(w, x[n]) via 4 chained f32 WMMAs (exact f32 math).
        v8f acc = {};
#pragma unroll
        for (int it = 0; it < 4; ++it)
            acc = __builtin_amdgcn_wmma_f32_16x16x4_f32(
                    false, wA[it], false, xb[it], (short)0, acc, false, false);

        // Per-lane partials over its 8 of the 16 elements of batch row n.
        float s_attn = 0.0f, s_q = 0.0f;
#pragma unroll
        for (int it = 0; it < 4; ++it) {
#pragma unroll
            for (int j = 0; j < 2; ++j) {
                float v = xb[it][j];
                s_attn += fast_sigmoid(aw * v);
                // p>0 ? sin^2(pi*p/2) : 0  ==  p>0 ? 0.5 - 0.5*cos(pi*p) : 0
                float qv = 0.5f - 0.5f * __cosf(PI * v);
                s_q += (v > 0.0f) ? qv : 0.0f;
            }
        }
        // Merge the two lane-halves (lane n and lane n+16 share batch row n).
        s_attn += __shfl_xor(s_attn, 16, 32);
        s_q    += __shfl_xor(s_q,    16, 32);

        float dot = acc[0];                       // D[*, n], replicated over M
        float cls = fast_sigmoid(dot + b0) * (s_attn * (1.0f / 16.0f));
        float q   = s_q * (1.0f / 16.0f);
        float res = al * cls + (1.0f - al) * q;

        if (hi == 0 && row < batch)
            __builtin_nontemporal_store(res, out + row);
    }
}

extern "C" void kernel_launch(void* const* d_in, const int* in_sizes, int n_in,
                              void* d_out, int out_size, void* d_ws, size_t ws_size,
                              hipStream_t stream)
{
    const float* x      = (const float*)d_in[0];
    const float* conv_w = (const float*)d_in[1];
    const float* conv_b = (const float*)d_in[2];
    const float* attn_w = (const float*)d_in[3];
    const float* alpha  = (const float*)d_in[4];
    float* out = (float*)d_out;

    int batch  = in_sizes[0] / 16;          // x is (B,1,4,4)
    int nTiles = (batch + 15) / 16;

    const int threads = 256;                 // 8 waves per block
    int blocksNeeded = (nTiles + 7) / 8;     // 1 tile per wave minimum
    int blocks = blocksNeeded < 4096 ? (blocksNeeded > 0 ? blocksNeeded : 1) : 4096;

    ConvEnhanced_fused_kernel<<<blocks, threads, 0, stream>>>(
        x, conv_w, conv_b, attn_w, alpha, out, batch);
}